// GAT_26645977105016
// MI455X (gfx1250) — compile-verified
//
#include <hip/hip_runtime.h>
#include <hip/hip_bf16.h>
#include <math.h>

typedef __attribute__((ext_vector_type(2))) float v2f;
typedef __attribute__((ext_vector_type(8))) float v8f;

// =====================================================================
// WMMA f32 GEMM: C[M,Nc] = A[M,K] @ W[K,Nc] (+ optional bias per col)
// wave32; each wave -> 16 rows x 64 cols (4 accumulators, A reused 4x)
// 8 waves / block -> 128 x 64 output tile per block.
// =====================================================================
__global__ __launch_bounds__(256)
void gat_wmma_gemm(const float* __restrict__ A, const float* __restrict__ W,
                   const float* __restrict__ bias, float* __restrict__ C,
                   int M, int K, int Nc)
{
  const int lane = threadIdx.x & 31;
  const int wave = threadIdx.x >> 5;
  const int rowTile = blockIdx.x * 8 + wave;
  if (rowTile * 16 >= M) return;               // wave-uniform, EXEC stays all-1s
  const int row0    = rowTile * 16;
  const int colBase = blockIdx.y * 64;

  const int m16   = lane & 15;
  const int khalf = (lane >> 4) * 2;           // 0 or 2

  const float* aRowP = A + (size_t)(row0 + m16) * K + khalf;
  const float* bColP = W + (size_t)khalf * Nc + (colBase + m16);

  v8f acc0 = {}, acc1 = {}, acc2 = {}, acc3 = {};
  for (int k = 0; k < K; k += 4) {
    v2f a; a.x = aRowP[k]; a.y = aRowP[k + 1];
    const float* bp = bColP + (size_t)k * Nc;
    v2f b0, b1, b2, b3;
    b0.x = bp[0];  b0.y = bp[Nc];
    b1.x = bp[16]; b1.y = bp[Nc + 16];
    b2.x = bp[32]; b2.y = bp[Nc + 32];
    b3.x = bp[48]; b3.y = bp[Nc + 48];
    acc0 = __builtin_amdgcn_wmma_f32_16x16x4_f32(false, a, false, b0, (short)0, acc0, false, false);
    acc1 = __builtin_amdgcn_wmma_f32_16x16x4_f32(false, a, false, b1, (short)0, acc1, false, false);
    acc2 = __builtin_amdgcn_wmma_f32_16x16x4_f32(false, a, false, b2, (short)0, acc2, false, false);
    acc3 = __builtin_amdgcn_wmma_f32_16x16x4_f32(false, a, false, b3, (short)0, acc3, false, false);
  }
  const int rOff = (lane >> 4) * 8;            // lanes 16-31 hold M=8..15
  const int col0 = colBase + m16;
  const float bv0 = bias ? bias[col0]      : 0.f;
  const float bv1 = bias ? bias[col0 + 16] : 0.f;
  const float bv2 = bias ? bias[col0 + 32] : 0.f;
  const float bv3 = bias ? bias[col0 + 48] : 0.f;
  #pragma unroll
  for (int r = 0; r < 8; ++r) {
    size_t rowp = (size_t)(row0 + r + rOff) * Nc;
    C[rowp + col0]      = acc0[r] + bv0;
    C[rowp + col0 + 16] = acc1[r] + bv1;
    C[rowp + col0 + 32] = acc2[r] + bv2;
    C[rowp + col0 + 48] = acc3[r] + bv3;
  }
}

// ============================ CSR construction ===============================
__global__ __launch_bounds__(256)
void deg_count(const int* __restrict__ ei, int E_, int Nn, int* __restrict__ deg)
{
  int e = blockIdx.x * blockDim.x + threadIdx.x;
  if (e >= E_ + Nn) return;
  int dst = (e < E_) ? ei[E_ + e] : (e - E_);   // implicit self-loops
  atomicAdd(&deg[dst], 1);
}

// Single-block exclusive scan: rowptr[0..n] from deg[0..n-1]. blockDim = 1024.
__global__ __launch_bounds__(1024)
void scan_single(const int* __restrict__ deg, int* __restrict__ rowptr, int n)
{
  __shared__ int part[1024];
  const int tid = threadIdx.x;
  const int per = (n + 1024) / 1024;            // covers idx 0..n inclusive
  const int base = tid * per;
  int s = 0;
  for (int i = 0; i < per; ++i) {
    int idx = base + i;
    if (idx < n) s += deg[idx];
  }
  part[tid] = s;
  __syncthreads();
  for (int off = 1; off < 1024; off <<= 1) {    // Hillis-Steele inclusive scan
    int add = (tid >= off) ? part[tid - off] : 0;
    __syncthreads();
    part[tid] += add;
    __syncthreads();
  }
  int run = (tid == 0) ? 0 : part[tid - 1];     // exclusive prefix of this chunk
  for (int i = 0; i < per; ++i) {
    int idx = base + i;
    if (idx <= n) rowptr[idx] = run;
    if (idx < n)  run += deg[idx];
  }
}

__global__ __launch_bounds__(256)
void csr_fill(const int* __restrict__ ei, int E_, int Nn,
              const int* __restrict__ rowptr, int* __restrict__ cursor,
              int* __restrict__ srcs)
{
  int e = blockIdx.x * blockDim.x + threadIdx.x;
  if (e >= E_ + Nn) return;
  int src = (e < E_) ? ei[e]      : (e - E_);
  int dst = (e < E_) ? ei[E_ + e] : (e - E_);
  int off = atomicAdd(&cursor[dst], 1);
  srcs[rowptr[dst] + off] = src;
}

// ---------- per-(node,head) attention scores: dot(h, a_s), dot(h, a_d) -------
__global__ __launch_bounds__(256)
void att_scores(const float* __restrict__ hfeat, const float* __restrict__ a_s,
                const float* __restrict__ a_d, float* __restrict__ sS,
                float* __restrict__ sD, int Nn, int Hh, int Cc)
{
  int i = blockIdx.x * blockDim.x + threadIdx.x;
  if (i >= Nn * Hh) return;
  int n = i / Hh, h = i % Hh;
  const float* p  = hfeat + ((size_t)n * Hh + h) * Cc;
  const float* as = a_s + h * Cc;
  const float* ad = a_d + h * Cc;
  float ss = 0.f, sd = 0.f;
  #pragma unroll 4
  for (int c = 0; c < Cc; ++c) { float v = p[c]; ss += v * as[c]; sd += v * ad[c]; }
  sS[i] = ss; sD[i] = sd;
}

// ---------- gather segment-softmax: alpha in CSR order, no atomics -----------
__global__ __launch_bounds__(256)
void seg_softmax(const int* __restrict__ rowptr, const int* __restrict__ srcs,
                 const float* __restrict__ sS, const float* __restrict__ sD,
                 int Nn, int Hh, float* __restrict__ alpha)
{
  int i = blockIdx.x * blockDim.x + threadIdx.x;
  if (i >= Nn * Hh) return;
  int d = i / Hh, h = i % Hh;
  int p0 = rowptr[d], p1 = rowptr[d + 1];
  float sdv = sD[d * Hh + h];
  float mx = -INFINITY;
  for (int p = p0; p < p1; ++p) {
    float l = sS[srcs[p] * Hh + h] + sdv;
    l = (l > 0.f) ? l : 0.2f * l;               // leaky_relu 0.2
    mx = fmaxf(mx, l);
  }
  float sum = 0.f;
  for (int p = p0; p < p1; ++p) {
    float l = sS[srcs[p] * Hh + h] + sdv;
    l = (l > 0.f) ? l : 0.2f * l;
    float ex = expf(l - mx);
    alpha[(size_t)p * Hh + h] = ex;
    sum += ex;
  }
  float inv = 1.f / (sum + 1e-16f);
  for (int p = p0; p < p1; ++p) alpha[(size_t)p * Hh + h] *= inv;
}

// ---------- gather aggregation: one block per dst, 4 channels per thread -----
// blockDim.x == Hh*Cc/4; each thread accumulates float4 in registers.
__global__ void gather_aggregate(const int* __restrict__ rowptr,
                                 const int* __restrict__ srcs,
                                 const float* __restrict__ alpha,
                                 const float* __restrict__ hfeat,
                                 float* __restrict__ out, int Hh, int Cc)
{
  const int d   = blockIdx.x;
  const int HCl = Hh * Cc;
  const int c0  = threadIdx.x * 4;
  const int h   = c0 / Cc;
  const int p0 = rowptr[d], p1 = rowptr[d + 1];
  float a0 = 0.f, a1 = 0.f, a2 = 0.f, a3 = 0.f;
  for (int p = p0; p < p1; ++p) {
    int src = srcs[p];
    float a = alpha[(size_t)p * Hh + h];
    const float4 hv = *(const float4*)(hfeat + (size_t)src * HCl + c0);
    a0 += a * hv.x; a1 += a * hv.y; a2 += a * hv.z; a3 += a * hv.w;
  }
  float4 r; r.x = a0; r.y = a1; r.z = a2; r.w = a3;
  *(float4*)(out + (size_t)d * HCl + c0) = r;
}

// ---------- BN stats: per-channel sum / sumsq (128 rows per block) -----------
__global__ __launch_bounds__(256)
void bn_partial(const float* __restrict__ x, int Nn, int Cch, float* __restrict__ sums)
{
  int r0 = blockIdx.x * 128;
  int r1 = (r0 + 128 < Nn) ? r0 + 128 : Nn;
  for (int j = threadIdx.x; j < Cch; j += blockDim.x) {
    float s = 0.f, s2 = 0.f;
    for (int n = r0; n < r1; ++n) {
      float v = x[(size_t)n * Cch + j];
      s += v; s2 += v * v;
    }
    atomicAdd(&sums[j], s);
    atomicAdd(&sums[Cch + j], s2);
  }
}

// ---------- fused BN -> (ELU) -> (+residual) ---------------------------------
__global__ __launch_bounds__(256)
void bn_elu_res(const float* __restrict__ x, const float* __restrict__ sums,
                const float* __restrict__ g, const float* __restrict__ be,
                const float* __restrict__ resid, float* __restrict__ out,
                int Nn, int Cch, int doElu, int doRes)
{
  int idx = blockIdx.x * blockDim.x + threadIdx.x;
  if (idx >= Nn * Cch) return;
  int c = idx % Cch;
  float mu  = sums[c] / (float)Nn;
  float var = sums[Cch + c] / (float)Nn - mu * mu;
  float y = g[c] * (x[idx] - mu) / sqrtf(var + 1e-5f) + be[c];
  if (doElu) y = (y > 0.f) ? y : expm1f(y);
  if (doRes) y += resid[idx];
  out[idx] = y;
}

// ---------- graph boundaries via binary search (batch is sorted) -------------
__global__ __launch_bounds__(128)
void graph_bounds(const int* __restrict__ batch, int Nn, int Gg, int* __restrict__ gb)
{
  int g = blockIdx.x * blockDim.x + threadIdx.x;
  if (g > Gg) return;
  int lo = 0, hi = Nn;                           // first index with batch[i] >= g
  while (lo < hi) { int mid = (lo + hi) >> 1; if (batch[mid] < g) lo = mid + 1; else hi = mid; }
  gb[g] = lo;
}

// ---------- per-graph mean pool: one block per graph, 1 channel / thread -----
__global__ void pool_mean(const float* __restrict__ x, const int* __restrict__ gb,
                          int Cch, float* __restrict__ outp)
{
  int g = blockIdx.x;
  int c = threadIdx.x;                           // blockDim == Cch
  int r0 = gb[g], r1 = gb[g + 1];
  float s = 0.f;
  for (int n = r0; n < r1; ++n) s += x[(size_t)n * Cch + c];
  float cnt = fmaxf((float)(r1 - r0), 1.f);
  outp[g * Cch + c] = s / cnt;
}

// =====================================================================
extern "C" void kernel_launch(void* const* d_in, const int* in_sizes, int n_in,
                              void* d_out, int out_size, void* d_ws, size_t ws_size,
                              hipStream_t stream)
{
  (void)in_sizes; (void)n_in; (void)out_size; (void)ws_size;
  const int N = 10000, E = 160000, H = 8, C = 64, HCv = 512, G = 64;
  const int NNZ = E + N;

  const float* x     = (const float*)d_in[0];
  const int*   ei    = (const int*)  d_in[1];
  const int*   batch = (const int*)  d_in[2];
  const float* encW  = (const float*)d_in[3];
  const float* encB  = (const float*)d_in[4];
  const float* W1    = (const float*)d_in[5];
  const float* as1   = (const float*)d_in[6];
  const float* ad1   = (const float*)d_in[7];
  // d_in[8]=b1, d_in[14]=b2, d_in[20]=b3 are added before BN -> cancel exactly
  const float* g1    = (const float*)d_in[9];
  const float* be1   = (const float*)d_in[10];
  const float* W2    = (const float*)d_in[11];
  const float* as2   = (const float*)d_in[12];
  const float* ad2   = (const float*)d_in[13];
  const float* g2    = (const float*)d_in[15];
  const float* be2   = (const float*)d_in[16];
  const float* W3    = (const float*)d_in[17];
  const float* as3   = (const float*)d_in[18];
  const float* ad3   = (const float*)d_in[19];
  const float* g3    = (const float*)d_in[21];
  const float* be3   = (const float*)d_in[22];
  const float* linW  = (const float*)d_in[23];
  const float* linB  = (const float*)d_in[24];

  float* ws = (float*)d_ws;
  const size_t NHC = (size_t)N * HCv;
  float* xA   = ws;                          // running activation [N,512]
  float* xF   = xA + NHC;                    // h = x @ W          [N,512]
  float* xO   = xF + NHC;                    // aggregation output [N,512]
  float* sS   = xO + NHC;                    // [N,H]
  float* sD   = sS + (size_t)N * H;
  float* alpha = sD + (size_t)N * H;         // CSR-ordered alpha [NNZ,H]
  float* bns  = alpha + (size_t)NNZ * H;     // BN sum/sumsq [2*512]
  float* pool = bns + 1024;                  // [G,C]
  int*   ipart  = (int*)(pool + G * C);
  int*   deg    = ipart;                     // [N]
  int*   rowptr = deg + N;                   // [N+1]
  int*   cursor = rowptr + (N + 1);          // [N]
  int*   srcs   = cursor + N;                // [NNZ]
  int*   gb     = srcs + NNZ;                // [G+1]

  dim3 blk(256);

  // ---- CSR over dst (built once, reused by all layers)
  hipMemsetAsync(deg, 0, (size_t)N * sizeof(int), stream);
  deg_count<<<(NNZ + 255) / 256, blk, 0, stream>>>(ei, E, N, deg);
  scan_single<<<1, 1024, 0, stream>>>(deg, rowptr, N);
  hipMemsetAsync(cursor, 0, (size_t)N * sizeof(int), stream);
  csr_fill<<<(NNZ + 255) / 256, blk, 0, stream>>>(ei, E, N, rowptr, cursor, srcs);
  graph_bounds<<<1, 128, 0, stream>>>(batch, N, G, gb);

  // ---- encoder: xA = x @ enc_W + enc_b
  {
    dim3 gg((N / 16 + 7) / 8, HCv / 64);
    gat_wmma_gemm<<<gg, blk, 0, stream>>>(x, encW, encB, xA, N, 512, HCv);
  }

  // ---- GAT layers 1 & 2 (concat heads, BN, ELU, residual)
  const float* Ws[2]  = {W1, W2};
  const float* asv[2] = {as1, as2};
  const float* adv[2] = {ad1, ad2};
  const float* gv[2]  = {g1, g2};
  const float* bev[2] = {be1, be2};
  for (int L = 0; L < 2; ++L) {
    dim3 gg((N / 16 + 7) / 8, HCv / 64);
    gat_wmma_gemm<<<gg, blk, 0, stream>>>(xA, Ws[L], nullptr, xF, N, HCv, HCv);

    int nh = N * H;
    att_scores<<<(nh + 255) / 256, blk, 0, stream>>>(xF, asv[L], adv[L], sS, sD, N, H, C);
    seg_softmax<<<(nh + 255) / 256, blk, 0, stream>>>(rowptr, srcs, sS, sD, N, H, alpha);
    gather_aggregate<<<N, HCv / 4, 0, stream>>>(rowptr, srcs, alpha, xF, xO, H, C);

    hipMemsetAsync(bns, 0, 2 * HCv * sizeof(float), stream);
    bn_partial<<<(N + 127) / 128, blk, 0, stream>>>(xO, N, HCv, bns);
    bn_elu_res<<<((N * HCv) + 255) / 256, blk, 0, stream>>>(xO, bns, gv[L], bev[L],
                                                            xA, xA, N, HCv, 1, 1);
  }

  // ---- layer 3 (H=1, C=64, BN, no ELU/residual)
  {
    dim3 gg((N / 16 + 7) / 8, 1);
    gat_wmma_gemm<<<gg, blk, 0, stream>>>(xA, W3, nullptr, xF, N, HCv, C);

    att_scores<<<(N + 255) / 256, blk, 0, stream>>>(xF, as3, ad3, sS, sD, N, 1, C);
    seg_softmax<<<(N + 255) / 256, blk, 0, stream>>>(rowptr, srcs, sS, sD, N, 1, alpha);
    gather_aggregate<<<N, C / 4, 0, stream>>>(rowptr, srcs, alpha, xF, xO, 1, C);

    hipMemsetAsync(bns, 0, 2 * C * sizeof(float), stream);
    bn_partial<<<(N + 127) / 128, blk, 0, stream>>>(xO, N, C, bns);
    bn_elu_res<<<((N * C) + 255) / 256, blk, 0, stream>>>(xO, bns, g3, be3,
                                                          nullptr, xO, N, C, 0, 0);

    // ---- global mean pool + classifier
    pool_mean<<<G, C, 0, stream>>>(xO, gb, C, pool);
    dim3 gl((G / 16 + 7) / 8, 1);
    gat_wmma_gemm<<<gl, blk, 0, stream>>>(pool, linW, linB, (float*)d_out, G, C, 64);
  }
}